// TensorTrainModel_75986561401083
// MI455X (gfx1250) — compile-verified
//
#include <hip/hip_runtime.h>
#include <hip/hip_bf16.h>
#include <stdint.h>

// Tensor-train forward: N x (M-step chain of 32x32 log-domain vec-mat products).
// Memory-bound: L is 1 GiB fp32 streamed once (~44us at 23.3 TB/s HBM).
// Strategy: one wave32 per sample, lane j owns column j (coalesced loads,
// lane-local k-reduction), async global->LDS double-buffered tile prefetch
// (ASYNCcnt) overlapping the exp-heavy VALU compute with the 4KB/tile DMA.
// WMMA is inapplicable: every sample has its own matrix exp(W_m + L[n,m]),
// so there is no shared-B GEMM, and the per-element exp dominates VALU work.

#define KDIM   32
#define MSTEPS 32
#define WAVES_PER_BLOCK 8

#if __has_builtin(__builtin_amdgcn_global_load_async_to_lds_b128)
#define HAS_ASYNC_LDS 1
#else
#define HAS_ASYNC_LDS 0
#endif

#if HAS_ASYNC_LDS
typedef int v4i_vs __attribute__((vector_size(16)));
typedef __attribute__((address_space(1))) v4i_vs* gbl_v4i_ptr;
typedef __attribute__((address_space(3))) v4i_vs* lds_v4i_ptr;

__device__ __forceinline__ void async_tile_copy(const float* g, float* l, int lane) {
    // 4 KB tile = 8 x (32 lanes x 16B) async b128 transfers, contiguous LDS.
    const float* gp = g + lane * 4;
    float*       lp = l + lane * 4;
#pragma unroll
    for (int c = 0; c < 8; ++c) {
        __builtin_amdgcn_global_load_async_to_lds_b128(
            (gbl_v4i_ptr)(uintptr_t)(gp + c * 128),
            (lds_v4i_ptr)(uint32_t)(uintptr_t)(lp + c * 128),
            0, 0);
    }
}

__device__ __forceinline__ void wait_async_le8() {
#if __has_builtin(__builtin_amdgcn_s_wait_asynccnt)
    __builtin_amdgcn_s_wait_asynccnt(8);
#else
    asm volatile("s_wait_asynccnt 0x8" ::: "memory");
#endif
    asm volatile("" ::: "memory");
}

__device__ __forceinline__ void wait_async_le0() {
#if __has_builtin(__builtin_amdgcn_s_wait_asynccnt)
    __builtin_amdgcn_s_wait_asynccnt(0);
#else
    asm volatile("s_wait_asynccnt 0x0" ::: "memory");
#endif
    asm volatile("" ::: "memory");
}
#endif // HAS_ASYNC_LDS

// c[m][j] = logsumexp_k W[m][k][j]  (column-wise over rows; 32 blocks x 32 lanes)
__global__ void tt_precompute_c(const float* __restrict__ W, float* __restrict__ c) {
    const int m = blockIdx.x;
    const int j = threadIdx.x;
    const float* Wm = W + m * KDIM * KDIM;
    float mx = -INFINITY;
#pragma unroll
    for (int k = 0; k < KDIM; ++k) mx = fmaxf(mx, Wm[k * KDIM + j]);
    float s = 0.f;
#pragma unroll
    for (int k = 0; k < KDIM; ++k) s += __expf(Wm[k * KDIM + j] - mx);
    c[m * KDIM + j] = mx + __logf(s);
}

__global__ __launch_bounds__(WAVES_PER_BLOCK * 32, 1)
void tt_forward(const float* __restrict__ L, const float* __restrict__ wk0,
                const float* __restrict__ W, const float* __restrict__ c,
                float* __restrict__ out, int Ntot) {
    const int lane = threadIdx.x & 31;
    const int wave = threadIdx.x >> 5;
    const int n    = blockIdx.x * WAVES_PER_BLOCK + wave;

#if HAS_ASYNC_LDS
    __shared__ float tiles[WAVES_PER_BLOCK][2][KDIM * KDIM]; // 64 KB, wave-private buffers
    float* buf0 = &tiles[wave][0][0];
    float* buf1 = &tiles[wave][1][0];
#endif

    if (n >= Ntot) return; // wave-uniform

    const float* Lbase = L + (size_t)n * MSTEPS * KDIM * KDIM;

    // a0[j] = wk0[j] - logsumexp(wk0)
    float w0 = wk0[lane];
    float mx = w0;
#pragma unroll
    for (int off = 16; off; off >>= 1) mx = fmaxf(mx, __shfl_xor(mx, off, 32));
    float s0 = __expf(w0 - mx);
#pragma unroll
    for (int off = 16; off; off >>= 1) s0 += __shfl_xor(s0, off, 32);
    float a = w0 - (mx + __logf(s0));

#if HAS_ASYNC_LDS
    async_tile_copy(Lbase, buf0, lane); // prefetch m=0
#endif

    for (int m = 0; m < MSTEPS; ++m) {
#if HAS_ASYNC_LDS
        // Kick off next tile's DMA first: it overlaps the reduction below
        // in addition to the previous iteration's whole inner loop.
        float* cur = (m & 1) ? buf1 : buf0;
        float* nxt = (m & 1) ? buf0 : buf1;
        const bool have_next = (m + 1 < MSTEPS);
        if (have_next) async_tile_copy(Lbase + (size_t)(m + 1) * KDIM * KDIM, nxt, lane);
#endif

        // amax over lanes; p = exp(a - amax)
        float amax = a;
#pragma unroll
        for (int off = 16; off; off >>= 1) amax = fmaxf(amax, __shfl_xor(amax, off, 32));
        float p = __expf(a - amax);

#if HAS_ASYNC_LDS
        if (have_next) wait_async_le8(); // newest 8 (next tile) stay in flight; current tile done
        else           wait_async_le0();
#endif

        const float* Wm = W + m * KDIM * KDIM;
        const float  cj = c[m * KDIM + lane];

        float acc = 0.f;
#pragma unroll
        for (int k = 0; k < KDIM; ++k) {
            const float pk = __shfl(p, k, 32); // constant k -> v_readlane
#if HAS_ASYNC_LDS
            const float lv = cur[k * KDIM + lane];                    // ds_load_b32, bank-conflict-free
#else
            const float lv = Lbase[(size_t)m * KDIM * KDIM + k * KDIM + lane];
#endif
            const float t = Wm[k * KDIM + lane] + lv; // W tile stays hot in WGP$/L2
            acc = fmaf(pk, __expf(t), acc);
        }
        a = amax - cj + __logf(acc);
    }

    // out[n] = logsumexp_j a[j]
    float fm = a;
#pragma unroll
    for (int off = 16; off; off >>= 1) fm = fmaxf(fm, __shfl_xor(fm, off, 32));
    float fs = __expf(a - fm);
#pragma unroll
    for (int off = 16; off; off >>= 1) fs += __shfl_xor(fs, off, 32);
    if (lane == 0) out[n] = fm + __logf(fs);
}

extern "C" void kernel_launch(void* const* d_in, const int* in_sizes, int n_in,
                              void* d_out, int out_size, void* d_ws, size_t ws_size,
                              hipStream_t stream) {
    const float* L   = (const float*)d_in[0];
    const float* wk0 = (const float*)d_in[1];
    const float* W   = (const float*)d_in[2];
    float* out = (float*)d_out;
    float* c   = (float*)d_ws; // MSTEPS*KDIM floats = 4 KB

    const int Ntot = in_sizes[0] / (MSTEPS * KDIM * KDIM);

    tt_precompute_c<<<MSTEPS, KDIM, 0, stream>>>(W, c);

    const int blocks = (Ntot + WAVES_PER_BLOCK - 1) / WAVES_PER_BLOCK;
    tt_forward<<<blocks, WAVES_PER_BLOCK * 32, 0, stream>>>(L, wk0, W, c, out, Ntot);
}